// RobustMedBN2d_45466523795512
// MI455X (gfx1250) — compile-verified
//
#include <hip/hip_runtime.h>
#include <stdint.h>

typedef unsigned int uint;
typedef __attribute__((ext_vector_type(2))) float v2f;
typedef __attribute__((ext_vector_type(8))) float v8f;

#define C_     128
#define B_     64
#define HW_    3136          // 56*56
#define NPC    200704        // B_*HW_
#define KRANK  100351        // (NPC-1)/2
#define NBIN   2048
#define NTILE  49            // 3136 / 64 floats per b64-tile per wave
#define MOM    0.1f
#define EPSV   1e-5f

// ---- monotone key <-> float -------------------------------------------------
__device__ __forceinline__ uint f2key(float v) {
  uint u = __float_as_uint(v);
  return (u & 0x80000000u) ? ~u : (u | 0x80000000u);
}
__device__ __forceinline__ float key2f(uint k) {
  uint u = (k & 0x80000000u) ? (k ^ 0x80000000u) : ~k;
  return __uint_as_float(u);
}

// ---- 32-lane f32 reduction using one V_WMMA_F32_16X16X4_F32 -----------------
// A: VGPR0 = partial (A[m][0]=p_m, A[m][2]=p_{m+16}), VGPR1 = 0. B = ones.
// D[m][n] = p_m + p_{m+16}; sum 8 accumulator VGPRs then xor-16 shuffle.
__device__ __forceinline__ float wave_reduce_f32(float p) {
  v2f a; a.x = p;    a.y = 0.0f;
  v2f b; b.x = 1.0f; b.y = 1.0f;
  v8f c = {};
  c = __builtin_amdgcn_wmma_f32_16x16x4_f32(false, a, false, b, (short)0, c,
                                            false, false);
  float t = c[0] + c[1] + c[2] + c[3] + c[4] + c[5] + c[6] + c[7];
  t += __shfl_xor(t, 16, 32);
  return t;
}

// ---- K0: zero workspace -----------------------------------------------------
__global__ void k_zero(uint* __restrict__ hist, float* __restrict__ sums,
                       uint* __restrict__ candCount) {
  uint i = blockIdx.x * 256u + threadIdx.x;
  if (i < (uint)(C_ * NBIN)) hist[i] = 0u;
  if (i < 2u * C_) sums[i] = 0.0f;
  if (i < (uint)C_) candCount[i] = 0u;
}

// ---- K1: histogram + sum/sumsq, async global->LDS 4-deep pipelined stream ---
// grid (C_, 8), block 256 (8 waves); wave w handles plane b = blockIdx.y*8 + w.
__global__ void k_stats(const float* __restrict__ x, uint* __restrict__ ghist,
                        float* __restrict__ gsum, float* __restrict__ gsumsq) {
  __shared__ uint  hist[NBIN];
  __shared__ float2 stage[8][4][32];     // 4-deep per-wave staging (8 KB)

  const int c    = blockIdx.x;
  const int tid  = threadIdx.x;
  const int w    = tid >> 5;
  const int lane = tid & 31;

  for (int i = tid; i < NBIN; i += 256) hist[i] = 0u;
  __syncthreads();

  const int b = blockIdx.y * 8 + w;
  const float* g = x + ((size_t)b * C_ + c) * HW_;

  const uint ldsb = (uint)(size_t)(&stage[w][0][0]) + (uint)lane * 8u;
  const uint voffl = (uint)lane * 8u;

  float sum = 0.0f, ssq = 0.0f;

  // prime the 4-deep pipeline: tiles 0..3
  #pragma unroll
  for (int t = 0; t < 4; ++t) {
    uint dst  = ldsb + (uint)t * 256u;
    uint voff = voffl + (uint)t * 256u;
    asm volatile("global_load_async_to_lds_b64 %0, %1, %2"
                 :: "v"(dst), "v"(voff), "s"(g) : "memory");
  }

#define PROC_TILE(T)                                        \
  do {                                                      \
    float2 v = stage[w][(T) & 3][lane];                     \
    atomicAdd(&hist[f2key(v.x) >> 21], 1u);                 \
    sum += v.x; ssq = fmaf(v.x, v.x, ssq);                  \
    atomicAdd(&hist[f2key(v.y) >> 21], 1u);                 \
    sum += v.y; ssq = fmaf(v.y, v.y, ssq);                  \
  } while (0)

  uint voffn = voffl + 4u * 256u;        // next issue offset (tile 4)
  for (int i = 0; i <= NTILE - 4; ++i) { // tiles 0..45
    asm volatile("s_wait_asynccnt 3" ::: "memory");
    PROC_TILE(i);
    if (i <= NTILE - 5) {                // issue tile i+4 into buf (i&3)
      uint dst = ldsb + (uint)(i & 3) * 256u;
      asm volatile("global_load_async_to_lds_b64 %0, %1, %2"
                   :: "v"(dst), "v"(voffn), "s"(g) : "memory");
      voffn += 256u;
    }
  }
  // peeled tail: tiles 46, 47, 48
  asm volatile("s_wait_asynccnt 2" ::: "memory");
  PROC_TILE(NTILE - 3);
  asm volatile("s_wait_asynccnt 1" ::: "memory");
  PROC_TILE(NTILE - 2);
  asm volatile("s_wait_asynccnt 0" ::: "memory");
  PROC_TILE(NTILE - 1);
#undef PROC_TILE

  // WMMA-based cross-lane reduction (uniform flow, EXEC all ones)
  float wsum = wave_reduce_f32(sum);
  float wssq = wave_reduce_f32(ssq);
  if (lane == 0) {
    atomicAdd(&gsum[c], wsum);
    atomicAdd(&gsumsq[c], wssq);
  }

  __syncthreads();
  for (int i = tid; i < NBIN; i += 256) {
    uint h = hist[i];
    if (h) atomicAdd(&ghist[(size_t)c * NBIN + i], h);
  }
}

// ---- K2: per-channel scan of 2048 bins -> selected bin + residual rank ------
// grid C_, block 32 (one wave)
__global__ void k_scan(const uint* __restrict__ ghist, uint* __restrict__ selBin,
                       uint* __restrict__ selRank) {
  const int c = blockIdx.x;
  const int lane = threadIdx.x;
  uint base = 0;
  for (int chunk = 0; chunk < NBIN / 32; ++chunk) {
    uint v = ghist[(size_t)c * NBIN + chunk * 32 + lane];
    uint p = v;                                   // inclusive wave prefix
    for (int d = 1; d < 32; d <<= 1) {
      uint t = __shfl_up(p, d, 32);
      if (lane >= d) p += t;
    }
    uint total = __shfl(p, 31, 32);
    bool hit = (base + p) > (uint)KRANK;
    unsigned long long m = __ballot(hit);
    if (m) {
      int first = __ffsll((long long)m) - 1;
      if (lane == first) {
        selBin[c]  = (uint)(chunk * 32 + lane);
        selRank[c] = (uint)KRANK - (base + p - v);
      }
      return;
    }
    base += total;
  }
  if (lane == 0) { selBin[c] = NBIN - 1; selRank[c] = 0; }  // unreachable
}

// ---- K3: compact candidate keys in the selected bin -------------------------
// grid (C_, 8), block 256; wave w handles plane b = blockIdx.y*8 + w.
__global__ void k_cand(const float* __restrict__ x, const uint* __restrict__ selBin,
                       uint* __restrict__ candCount, uint* __restrict__ cand) {
  const int c    = blockIdx.x;
  const int w    = threadIdx.x >> 5;
  const int lane = threadIdx.x & 31;
  const int b    = blockIdx.y * 8 + w;
  const float2* g2 = (const float2*)(x + ((size_t)b * C_ + c) * HW_);
  const uint sb = selBin[c];

  for (int i = 0; i < NTILE; ++i) {
    float2 v = g2[i * 32 + lane];
    uint k0 = f2key(v.x), k1 = f2key(v.y);
    if ((k0 >> 21) == sb) {
      uint p = atomicAdd(&candCount[c], 1u);
      if (p < 2048u) cand[(size_t)c * 2048 + p] = k0;
    }
    if ((k1 >> 21) == sb) {
      uint p = atomicAdd(&candCount[c], 1u);
      if (p < 2048u) cand[(size_t)c * 2048 + p] = k1;
    }
  }
}

// ---- K4: exact rank select among candidates + finalize scale/shift ----------
__global__ void k_select(const uint* __restrict__ candCount,
                         const uint* __restrict__ selRank,
                         const uint* __restrict__ cand,
                         const float* __restrict__ gsum,
                         const float* __restrict__ gsumsq,
                         const float* __restrict__ wgt,
                         const float* __restrict__ bias,
                         const float* __restrict__ smean,
                         const float* __restrict__ svar,
                         float* __restrict__ scaleArr,
                         float* __restrict__ shiftArr) {
  __shared__ uint  sk[2048];
  __shared__ float smed;
  const int c = blockIdx.x;
  const int t = threadIdx.x;

  uint n = candCount[c];
  if (n > 2048u) n = 2048u;
  const uint r = selRank[c];
  if (t == 0) smed = 0.0f;
  for (uint j = t; j < n; j += 256) sk[j] = cand[(size_t)c * 2048 + j];
  __syncthreads();

  for (uint j = t; j < n; j += 256) {
    uint k = sk[j], less = 0, eq = 0;
    for (uint i = 0; i < n; ++i) {
      uint ki = sk[i];
      less += (ki < k);
      eq   += (ki == k);
    }
    if (less <= r && r < less + eq) smed = key2f(k);  // unique value, benign race
  }
  __syncthreads();

  if (t == 0) {
    const float m    = smed;
    const float invn = 1.0f / (float)NPC;
    const float my   = gsum[c] * invn;
    const float msq  = gsumsq[c] * invn;
    const float bvar = msq - 2.0f * m * my + m * m;
    const float mean = (1.0f - MOM) * smean[c] + MOM * m;
    const float var  = (1.0f - MOM) * svar[c]  + MOM * bvar;
    const float inv  = rsqrtf(var + EPSV);
    const float sc   = inv * wgt[c];
    scaleArr[c] = sc;
    shiftArr[c] = bias[c] - mean * sc;
  }
}

// ---- K5: out = fma(x, scale[c], shift[c]) -----------------------------------
// grid 25088, block 256; one float4 per thread. HW_=3136 -> 784 float4/plane.
__global__ void k_norm(const float4* __restrict__ x4,
                       const float* __restrict__ scaleArr,
                       const float* __restrict__ shiftArr,
                       float4* __restrict__ out4) {
  __shared__ float s_sc[C_];
  __shared__ float s_sh[C_];
  const int t = threadIdx.x;
  if (t < C_) s_sc[t] = scaleArr[t];
  else        s_sh[t - C_] = shiftArr[t - C_];
  __syncthreads();

  uint i4 = blockIdx.x * 256u + (uint)t;
  uint p  = i4 / 784u;          // plane id
  uint c  = p & (C_ - 1);       // plane % 128
  float4 v = x4[i4];
  float sc = s_sc[c], sh = s_sh[c];
  float4 o;
  o.x = fmaf(v.x, sc, sh);
  o.y = fmaf(v.y, sc, sh);
  o.z = fmaf(v.z, sc, sh);
  o.w = fmaf(v.w, sc, sh);
  out4[i4] = o;
}

extern "C" void kernel_launch(void* const* d_in, const int* in_sizes, int n_in,
                              void* d_out, int out_size, void* d_ws, size_t ws_size,
                              hipStream_t stream) {
  const float* x      = (const float*)d_in[0];
  const float* weight = (const float*)d_in[1];
  const float* bias   = (const float*)d_in[2];
  const float* smean  = (const float*)d_in[3];
  const float* svar   = (const float*)d_in[4];
  float* out = (float*)d_out;

  uint8_t* ws = (uint8_t*)d_ws;
  uint*  hist      = (uint*)ws;                              // 1 MB
  uint*  cand      = (uint*)(ws + (1u << 20));               // 1 MB
  float* sums      = (float*)(ws + (2u << 20));              // 256 f32: sum|sumsq
  uint*  candCount = (uint*)(ws + (2u << 20) + 1024);        // 128 u32
  uint*  selBin    = (uint*)(ws + (2u << 20) + 1536);        // 128 u32
  uint*  selRank   = (uint*)(ws + (2u << 20) + 2048);        // 128 u32
  float* scaleArr  = (float*)(ws + (2u << 20) + 2560);       // 128 f32
  float* shiftArr  = (float*)(ws + (2u << 20) + 3072);       // 128 f32

  dim3 gcs(C_, 8);
  k_zero<<<(C_ * NBIN) / 256, 256, 0, stream>>>(hist, sums, candCount);
  k_stats<<<gcs, 256, 0, stream>>>(x, hist, sums, sums + C_);
  k_scan<<<C_, 32, 0, stream>>>(hist, selBin, selRank);
  k_cand<<<gcs, 256, 0, stream>>>(x, selBin, candCount, cand);
  k_select<<<C_, 256, 0, stream>>>(candCount, selRank, cand, sums, sums + C_,
                                   weight, bias, smean, svar, scaleArr, shiftArr);
  k_norm<<<NPC * C_ / (256 * 4), 256, 0, stream>>>((const float4*)x, scaleArr,
                                                   shiftArr, (float4*)out);
}